// NCDcompReconstructor_87342454931803
// MI455X (gfx1250) — compile-verified
//
#include <hip/hip_runtime.h>
#include <math.h>

// ---------------------------------------------------------------------------
// Adjoint NUFFT (MRI recon) for MI455X / gfx1250.
//   1) KB gridding: scatter with global_atomic_add_f32 into L2-resident grid
//   2) 2-D inverse DFT as two complex GEMMs via V_WMMA_F32_16X16X4_F32,
//      restricted to the 320x320 output window that survives the crops.
//      Twiddles precomputed in WMMA-fragment order -> lean inner loops.
//   3) fused deapodization + magnitude epilogue
// ---------------------------------------------------------------------------

typedef __attribute__((ext_vector_type(2))) float v2f;
typedef __attribute__((ext_vector_type(8))) float v8f;

constexpr int B_   = 4;
constexpr int M_   = 262144;
constexpr int GX   = 1280;          // grid x
constexpr int GY   = 948;           // grid y
constexpr int ON   = 320;           // output rows/cols kept
constexpr int KTY  = 237;           // 948/4 k-steps in pass 1
constexpr int KTX  = 320;           // 1280/4 k-steps in pass 2
constexpr float PI_F     = 3.14159265358979323846f;
constexpr float BETA_F   = 13.8551004f;          // pi*sqrt(4.5^2-0.8)
constexpr float BETA2_F  = 191.96380560f;        // BETA^2
constexpr float PI6_F    = 18.8495559215f;       // pi*J, J=6
constexpr float CANG_Y   = 6.6278327e-3f;        // 2*pi/948
constexpr float CANG_X   = 4.9087385e-3f;        // 2*pi/1280
constexpr float SCALE_F  = 9.0780045e-4f;        // 1/sqrt(GX*GY)

__device__ __forceinline__ v8f wmma4(v2f a, v2f b, v8f c) {
  // D = A(16x4) * B(4x16) + C ; lowers to v_wmma_f32_16x16x4_f32
  return __builtin_amdgcn_wmma_f32_16x16x4_f32(
      false, a, false, b, (short)0, c, false, false);
}

// Abramowitz-Stegun modified Bessel I0 (f32), arg range here: [0, 13.856]
__device__ __forceinline__ float bessel_i0(float x) {
  if (x < 3.75f) {
    float t = x * x * (1.0f / (3.75f * 3.75f));
    return 1.0f + t * (3.5156229f + t * (3.0899424f + t * (1.2067492f +
               t * (0.2659732f + t * (0.0360768f + t * 0.0045813f)))));
  }
  float t = 3.75f / x;
  float p = 0.39894228f + t * (0.01328592f + t * (0.00225319f +
            t * (-0.00157565f + t * (0.00916281f + t * (-0.02057706f +
            t * (0.02635537f + t * (-0.01647633f + t * 0.00392377f)))))));
  return p * __expf(x) * rsqrtf(x);
}

// --------------------------- workspace zero --------------------------------
__global__ void __launch_bounds__(256) zero_ws(float4* __restrict__ p, int n4) {
  int i = blockIdx.x * blockDim.x + threadIdx.x;
  int stride = gridDim.x * blockDim.x;
  float4 z = {0.f, 0.f, 0.f, 0.f};
  for (; i < n4; i += stride) p[i] = z;
}

// ------------------- twiddle tables in WMMA B/A fragment order -------------
// Pass-1 B operand: Wy[k=y][n=c] = exp(2i*pi*y*yu(c)/948) / sqrt(GX*GY)
// entry index = (kt*20 + nt)*32 + lane ; per entry a v2f {k=k0, k=k0+1}
__global__ void __launch_bounds__(256) make_wy(float* __restrict__ WyR,
                                               float* __restrict__ WyI) {
  int id = blockIdx.x * blockDim.x + threadIdx.x;
  if (id >= KTY * 20 * 32) return;
  int lane = id & 31;
  int t2   = id >> 5;
  int nt   = t2 % 20;
  int kt   = t2 / 20;
  int ln = lane & 15, kh = lane >> 4;
  int c  = nt * 16 + ln;
  int yu = 788 + c; yu -= (yu >= GY) ? GY : 0;
  int y0 = 4 * kt + 2 * kh;
  unsigned p0 = (unsigned)(yu * y0) % (unsigned)GY;
  unsigned p1 = p0 + (unsigned)yu; p1 -= (p1 >= (unsigned)GY) ? GY : 0u;
  float s0, c0, s1, c1;
  __sincosf((float)p0 * CANG_Y, &s0, &c0);
  __sincosf((float)p1 * CANG_Y, &s1, &c1);
  WyR[2 * id + 0] = c0 * SCALE_F;  WyR[2 * id + 1] = c1 * SCALE_F;
  WyI[2 * id + 0] = s0 * SCALE_F;  WyI[2 * id + 1] = s1 * SCALE_F;
}

// Pass-2 A operand: Wx[m=r][k] = exp(2i*pi*k*xu(r)/1280)
// entry index = (mt*320 + kt)*32 + lane
__global__ void __launch_bounds__(256) make_wx(float* __restrict__ WxR,
                                               float* __restrict__ WxI) {
  int id = blockIdx.x * blockDim.x + threadIdx.x;
  if (id >= 20 * KTX * 32) return;
  int lane = id & 31;
  int t2   = id >> 5;
  int kt   = t2 % KTX;
  int mt   = t2 / KTX;
  int ln = lane & 15, kh = lane >> 4;
  int m  = mt * 16 + ln;
  int xu = 1120 + m; xu -= (xu >= GX) ? GX : 0;
  int k0 = 4 * kt + 2 * kh;
  unsigned p0 = (unsigned)(xu * k0) % (unsigned)GX;
  unsigned p1 = p0 + (unsigned)xu; p1 -= (p1 >= (unsigned)GX) ? GX : 0u;
  float s0, c0, s1, c1;
  __sincosf((float)p0 * CANG_X, &s0, &c0);
  __sincosf((float)p1 * CANG_X, &s1, &c1);
  WxR[2 * id + 0] = c0;  WxR[2 * id + 1] = c1;
  WxI[2 * id + 0] = s0;  WxI[2 * id + 1] = s1;
}

// --------------------------- KB gridding (scatter) -------------------------
__global__ void __launch_bounds__(256) grid_scatter(
    const float* __restrict__ kspace,   // [B][M][2]
    const float* __restrict__ traj,     // [B][2][M]
    const float* __restrict__ dcomp,    // [B][M]
    float* __restrict__ Gr, float* __restrict__ Gi) {
  int s = blockIdx.x * blockDim.x + threadIdx.x;
  if (s >= B_ * M_) return;
  int b = s >> 18;               // / M_
  int m = s & (M_ - 1);
  float gx = traj[(b * 2 + 0) * M_ + m] * ((float)GX / (2.0f * PI_F));
  float gy = traj[(b * 2 + 1) * M_ + m] * ((float)GY / (2.0f * PI_F));
  float dc = dcomp[s];
  float yr = kspace[2 * s + 0] * dc;
  float yi = kspace[2 * s + 1] * dc;

  int ixb = (int)floorf(gx - 3.0f) + 1;
  int iyb = (int)floorf(gy - 3.0f) + 1;
  float wx[6], wy[6];
  int iym[6];
#pragma unroll
  for (int o = 0; o < 6; ++o) {
    float dx = gx - (float)(ixb + o);
    float dy = gy - (float)(iyb + o);
    float tx = fmaxf(1.0f - dx * dx * (1.0f / 9.0f), 0.0f);   // (2d/J)^2
    float ty = fmaxf(1.0f - dy * dy * (1.0f / 9.0f), 0.0f);
    wx[o] = bessel_i0(BETA_F * sqrtf(tx));
    wy[o] = bessel_i0(BETA_F * sqrtf(ty));
    int iy = iyb + o;
    iy += (iy < 0) ? GY : 0;
    iy -= (iy >= GY) ? GY : 0;
    iym[o] = iy;
  }
  int base_b = b * GX * GY;
#pragma unroll
  for (int ox = 0; ox < 6; ++ox) {
    int ix = ixb + ox;
    ix += (ix < 0) ? GX : 0;
    ix -= (ix >= GX) ? GX : 0;
    int rowbase = base_b + ix * GY;
    float wxx = wx[ox];
#pragma unroll
    for (int oy = 0; oy < 6; ++oy) {
      float w = wxx * wy[oy];
      int a = rowbase + iym[oy];
      atomicAdd(&Gr[a], yr * w);   // global_atomic_add_f32
      atomicAdd(&Gi[a], yi * w);
    }
  }
}

// ---- Pass 1: T[b][c][x] (c-major!) = sum_y G[b][x][y]*Wy[y][c]
// One wave per 16x16 complex tile. Inner loop: 4 b64 loads + 4 WMMA.
__global__ void __launch_bounds__(256) dft_y(
    const float* __restrict__ Gr, const float* __restrict__ Gi,
    const float* __restrict__ WyR, const float* __restrict__ WyI,
    float* __restrict__ Trt, float* __restrict__ Tit) {
  int wid  = (blockIdx.x * blockDim.x + threadIdx.x) >> 5;
  int lane = threadIdx.x & 31;
  int nt = wid % 20;        // 320/16 column tiles
  int t2 = wid / 20;
  int mt = t2 % 80;         // 1280/16 row tiles
  int b  = t2 / 80;
  int ln = lane & 15;
  int kh = lane >> 4;

  // A fragment: lane row m = ln, k-pair chosen by kh
  size_t rowoff = ((size_t)(b * GX + mt * 16 + ln)) * GY + (size_t)(2 * kh);
  const float* arp = Gr + rowoff;
  const float* aip = Gi + rowoff;
  // B fragment table: stride 20*64 floats per kt
  const float* brp = WyR + (size_t)(nt * 32 + lane) * 2;
  const float* bip = WyI + (size_t)(nt * 32 + lane) * 2;

  v8f cr = {};
  v8f ci = {};
  for (int kt = 0; kt < KTY; ++kt) {
    v2f ar = *(const v2f*)(arp + 4 * kt);
    v2f ai = *(const v2f*)(aip + 4 * kt);
    v2f br = *(const v2f*)(brp + 1280 * kt);   // 20*32*2 floats per kt
    v2f bi = *(const v2f*)(bip + 1280 * kt);
    v2f bni = {-bi.x, -bi.y};
    cr = wmma4(ar, br,  cr);     // Re += Gr*cos
    cr = wmma4(ai, bni, cr);     // Re -= Gi*sin
    ci = wmma4(ar, bi,  ci);     // Im += Gr*sin
    ci = wmma4(ai, br,  ci);     // Im += Gi*cos
  }
  // D: vgpr v -> row x = mt*16 + v + 8*kh, col c = nt*16 + ln.
  // Store c-major so pass-2 B loads are contiguous v2f.
  int c = nt * 16 + ln;
  size_t base = ((size_t)(b * ON + c)) * GX + (size_t)(mt * 16 + 8 * kh);
  float4 r0 = {cr[0], cr[1], cr[2], cr[3]};
  float4 r1 = {cr[4], cr[5], cr[6], cr[7]};
  float4 i0v = {ci[0], ci[1], ci[2], ci[3]};
  float4 i1v = {ci[4], ci[5], ci[6], ci[7]};
  *(float4*)(Trt + base)     = r0;
  *(float4*)(Trt + base + 4) = r1;
  *(float4*)(Tit + base)     = i0v;
  *(float4*)(Tit + base + 4) = i1v;
}

// ---- Pass 2: O[r][c] = sum_k Wx[r][k] * T[k][c], fused deapod + |.|
__global__ void __launch_bounds__(256) dft_x_finalize(
    const float* __restrict__ Trt, const float* __restrict__ Tit,
    const float* __restrict__ WxR, const float* __restrict__ WxI,
    float* __restrict__ out) {
  int wid  = (blockIdx.x * blockDim.x + threadIdx.x) >> 5;
  int lane = threadIdx.x & 31;
  int nt = wid % 20;
  int t2 = wid / 20;
  int mt = t2 % 20;
  int b  = t2 / 20;
  int ln = lane & 15;
  int kh = lane >> 4;

  int c = nt * 16 + ln;   // B column / D column
  // A fragment table: stride 64 floats per kt
  const float* arp = WxR + ((size_t)mt * KTX * 64) + (size_t)lane * 2;
  const float* aip = WxI + ((size_t)mt * KTX * 64) + (size_t)lane * 2;
  // B fragment: contiguous rows x = 4*kt + 2*kh + {0,1} in c-major T
  const float* brp = Trt + ((size_t)(b * ON + c)) * GX + (size_t)(2 * kh);
  const float* bip = Tit + ((size_t)(b * ON + c)) * GX + (size_t)(2 * kh);

  v8f dr = {};
  v8f di = {};
  for (int kt = 0; kt < KTX; ++kt) {
    v2f ar = *(const v2f*)(arp + 64 * kt);
    v2f ai = *(const v2f*)(aip + 64 * kt);
    v2f br = *(const v2f*)(brp + 4 * kt);
    v2f bi = *(const v2f*)(bip + 4 * kt);
    v2f ani = {-ai.x, -ai.y};
    dr = wmma4(ar,  br, dr);     // Re += cos*Tr
    dr = wmma4(ani, bi, dr);     // Re -= sin*Ti
    di = wmma4(ar,  bi, di);     // Im += cos*Ti
    di = wmma4(ai,  br, di);     // Im += sin*Tr
  }

  // Deapodization (always on the sinh branch inside the crop window)
  float uy   = (float)(c - 160) * (1.0f / 948.0f);
  float argy = BETA2_F - (PI6_F * uy) * (PI6_F * uy);
  float sy   = sqrtf(argy);
  float dpy  = sy / sinhf(sy);
#pragma unroll
  for (int v = 0; v < 8; ++v) {
    int r = mt * 16 + v + 8 * kh;
    float ux   = (float)(r - 160) * (1.0f / 1280.0f);
    float argx = BETA2_F - (PI6_F * ux) * (PI6_F * ux);
    float sx   = sqrtf(argx);
    float dpx  = sx / sinhf(sx);
    float mag  = sqrtf(dr[v] * dr[v] + di[v] * di[v]);
    out[((size_t)b * ON + r) * ON + c] = mag * dpx * dpy;
  }
}

// ---------------------------------------------------------------------------
extern "C" void kernel_launch(void* const* d_in, const int* in_sizes, int n_in,
                              void* d_out, int out_size, void* d_ws, size_t ws_size,
                              hipStream_t stream) {
  (void)in_sizes; (void)n_in; (void)out_size;
  const float* kspace = (const float*)d_in[0];   // [4][262144][2]
  const float* traj   = (const float*)d_in[1];   // [4][2][262144]
  const float* dcomp  = (const float*)d_in[2];   // [4][262144]
  float* out = (float*)d_out;                    // [4][320][320]

  const size_t GSZ  = (size_t)B_ * GX * GY;      // 4,853,760 floats / plane
  const size_t TSZ  = (size_t)B_ * GX * ON;      // 1,638,400 floats / plane
  const size_t WYSZ = (size_t)KTY * 20 * 32 * 2; //   303,360 floats / plane
  const size_t WXSZ = (size_t)20 * KTX * 32 * 2; //   409,600 floats / plane
  const size_t need = (2 * GSZ + 2 * TSZ + 2 * WYSZ + 2 * WXSZ) * sizeof(float);
  if (ws_size < need) return;                    // ~57.6 MB

  float* Gr  = (float*)d_ws;
  float* Gi  = Gr  + GSZ;
  float* Trt = Gi  + GSZ;
  float* Tit = Trt + TSZ;
  float* WyR = Tit + TSZ;
  float* WyI = WyR + WYSZ;
  float* WxR = WyI + WYSZ;
  float* WxI = WxR + WXSZ;

  zero_ws<<<2048, 256, 0, stream>>>((float4*)Gr, (int)((2 * GSZ) / 4));
  make_wy<<<(KTY * 20 * 32 + 255) / 256, 256, 0, stream>>>(WyR, WyI);
  make_wx<<<(20 * KTX * 32) / 256, 256, 0, stream>>>(WxR, WxI);
  grid_scatter<<<(B_ * M_) / 256, 256, 0, stream>>>(kspace, traj, dcomp, Gr, Gi);
  dft_y<<<800, 256, 0, stream>>>(Gr, Gi, WyR, WyI, Trt, Tit);       // 6400 waves
  dft_x_finalize<<<200, 256, 0, stream>>>(Trt, Tit, WxR, WxI, out); // 1600 waves
}